// GATTrain_35021163331753
// MI455X (gfx1250) — compile-verified
//
#include <hip/hip_runtime.h>
#include <hip/hip_bf16.h>

typedef __attribute__((ext_vector_type(2))) float v2f;
typedef __attribute__((ext_vector_type(8))) float v8f;

#define NNODES 100000
#define NEDGES 1600000
#define INDIM  256
#define HDIM   64
#define NCLS   40
#define NTILES (NNODES / 16)   // 6250 exact 16-row tiles
#define NPAIRS (NTILES / 2)    // 3125 exact 32-row wave blocks
#define WSTRIDE 68             // padded LDS stride: conflict-free B loads

__device__ __forceinline__ v8f wmma_f32(v2f a, v2f b, v8f c) {
  // D(16x16) = A(16x4) * B(4x16) + C ; native f32 WMMA on gfx1250
  return __builtin_amdgcn_wmma_f32_16x16x4_f32(false, a, false, b,
                                               (short)0, c, false, false);
}

__device__ __forceinline__ void atomicMaxFloat(float* addr, float v) {
  // valid when initialized to -inf; positive path via signed int max,
  // negative path via unsigned int min (monotone float<->bits mappings)
  if (v >= 0.0f) atomicMax((int*)addr, __float_as_int(v));
  else           atomicMin((unsigned int*)addr, __float_as_uint(v));
}

// ---------------------------------------------------------------------------
// out[N,64] = in[N,K] @ W[K,64] + bias ; K multiple of 64 (256 or 64).
// 8 waves/block; each wave owns a 32x64 output block (2 M-tiles) so every
// LDS B-fragment feeds two WMMAs. W staged transposed (sWt[c][k], stride 68)
// so a B fragment is one aligned ds_load_b64.
// ---------------------------------------------------------------------------
__global__ __launch_bounds__(256) void gemm_feat(const float* __restrict__ in,
                                                 const float* __restrict__ W,
                                                 const float* __restrict__ bias,
                                                 float* __restrict__ out,
                                                 int K) {
  __shared__ float sWt[64 * WSTRIDE];
  const int t    = threadIdx.x;
  const int wave = t >> 5;
  const int lane = t & 31;
  const int n    = lane & 15;   // B col within 16-tile / A row within tile
  const int kh   = lane >> 4;   // K half-select for f32 WMMA fragments

  long pair  = (long)blockIdx.x * 8 + wave;
  bool valid = pair < NPAIRS;
  long row0  = (valid ? pair : (NPAIRS - 1)) * 32;
  const float* arow0 = in + (row0 + n) * (long)K;
  const float* arow1 = in + (row0 + 16 + n) * (long)K;

  v8f   acc[2][4];
  float bcol[4];
#pragma unroll
  for (int ct = 0; ct < 4; ++ct) {
    bcol[ct] = bias[ct * 16 + n];
#pragma unroll
    for (int i = 0; i < 8; ++i) { acc[0][ct][i] = 0.0f; acc[1][ct][i] = 0.0f; }
  }

  for (int kb = 0; kb < K; kb += 64) {
    __syncthreads();
    const float* Wc = W + (long)kb * 64;   // contiguous 4096-float slab
#pragma unroll
    for (int i = 0; i < 16; ++i) {
      int idx = t + i * 256;               // coalesced read, transposed write
      sWt[(idx & 63) * WSTRIDE + (idx >> 6)] = Wc[idx];
    }
    __syncthreads();
#pragma unroll
    for (int kk = 0; kk < 16; ++kk) {
      const int k0 = kk * 4 + kh * 2;
      v2f a0 = *(const v2f*)(arow0 + kb + k0);
      v2f a1 = *(const v2f*)(arow1 + kb + k0);
#pragma unroll
      for (int ct = 0; ct < 4; ++ct) {
        v2f b = *(const v2f*)(&sWt[(ct * 16 + n) * WSTRIDE + k0]);
        acc[0][ct] = wmma_f32(a0, b, acc[0][ct]);
        acc[1][ct] = wmma_f32(a1, b, acc[1][ct]);
      }
    }
  }

  if (valid) {
#pragma unroll
    for (int mt = 0; mt < 2; ++mt)
#pragma unroll
      for (int ct = 0; ct < 4; ++ct)
#pragma unroll
        for (int i = 0; i < 8; ++i) {
          long r = row0 + mt * 16 + i + 8 * kh;  // C/D: VGPR i -> M=i / M=i+8
          out[r * 64 + ct * 16 + n] = acc[mt][ct][i] + bcol[ct];
        }
  }
}

// ---------------------------------------------------------------------------
// Per-node: sdot = h . aw[:64], ddot = h . aw[64:]; also init emax/denom/agg.
// ---------------------------------------------------------------------------
__global__ __launch_bounds__(256) void attn_dots(const float* __restrict__ h,
                                                 const float* __restrict__ aw,
                                                 float* __restrict__ sdot,
                                                 float* __restrict__ ddot,
                                                 float* __restrict__ emax,
                                                 float* __restrict__ denom,
                                                 float* __restrict__ agg) {
  int i = blockIdx.x * 256 + threadIdx.x;
  if (i >= NNODES) return;
  float s = 0.0f, d = 0.0f;
  long base = (long)i * 64;
#pragma unroll 8
  for (int c = 0; c < 64; ++c) {
    float v = h[base + c];
    s += v * aw[c];
    d += v * aw[64 + c];
    agg[base + c] = 0.0f;
  }
  sdot[i]  = s;
  ddot[i]  = d;
  emax[i]  = -__builtin_huge_valf();
  denom[i] = 0.0f;
}

// ---------------------------------------------------------------------------
// Per-edge logits: e = leaky_relu(sdot[src] + ddot[dst]); segment max -> emax.
// ---------------------------------------------------------------------------
__global__ __launch_bounds__(256) void edge_logits(const int* __restrict__ src,
                                                   const int* __restrict__ dst,
                                                   const float* __restrict__ sdot,
                                                   const float* __restrict__ ddot,
                                                   float* __restrict__ e,
                                                   float* __restrict__ emax) {
  int i = blockIdx.x * 256 + threadIdx.x;
  if (i >= NEDGES) return;
  int s = src[i], d = dst[i];
  float v = sdot[s] + ddot[d];
  v = v > 0.0f ? v : 0.01f * v;
  e[i] = v;
  atomicMaxFloat(&emax[d], v);
}

// ---------------------------------------------------------------------------
// Per-edge: ee = exp(e - emax[dst]); segment sum -> denom. ee stored in place.
// ---------------------------------------------------------------------------
__global__ __launch_bounds__(256) void edge_exp(const int* __restrict__ dst,
                                                const float* __restrict__ emax,
                                                float* __restrict__ e,
                                                float* __restrict__ denom) {
  int i = blockIdx.x * 256 + threadIdx.x;
  if (i >= NEDGES) return;
  int d = dst[i];
  float ee = __expf(e[i] - emax[d]);
  e[i] = ee;
  unsafeAtomicAdd(&denom[d], ee);
}

// ---------------------------------------------------------------------------
// Per-edge aggregation: agg[dst] += (ee/denom[dst]) * h[src]. One wave32 per
// edge; each lane handles 2 of the 64 feature columns (float2 gather).
// ---------------------------------------------------------------------------
__global__ __launch_bounds__(256) void edge_agg(const int* __restrict__ src,
                                                const int* __restrict__ dst,
                                                const float* __restrict__ ee,
                                                const float* __restrict__ denom,
                                                const float* __restrict__ h,
                                                float* __restrict__ agg) {
  int  t    = threadIdx.x;
  long eidx = (long)blockIdx.x * 8 + (t >> 5);
  if (eidx >= NEDGES) return;
  int lane = t & 31;
  int s = src[eidx], d = dst[eidx];
  float alpha = ee[eidx] / denom[d];
  v2f hv = *(const v2f*)(h + (long)s * 64 + lane * 2);
  float* ap = agg + (long)d * 64 + lane * 2;
  unsafeAtomicAdd(ap,     alpha * hv.x);
  unsafeAtomicAdd(ap + 1, alpha * hv.y);
}

// ---------------------------------------------------------------------------
// Node update: out = relu(h @ gw[:64] + (agg - h) @ gw[64:] + gb). Two WMMA
// chains share the same A-fragment loads (h and agg-h at identical (row,k)).
// ---------------------------------------------------------------------------
__global__ __launch_bounds__(256) void gemm_update(const float* __restrict__ h,
                                                   const float* __restrict__ agg,
                                                   const float* __restrict__ gw,
                                                   const float* __restrict__ gb,
                                                   float* __restrict__ out) {
  __shared__ float sWA[64 * WSTRIDE];
  __shared__ float sWB[64 * WSTRIDE];
  const int t    = threadIdx.x;
  const int wave = t >> 5;
  const int lane = t & 31;
  const int n    = lane & 15;
  const int kh   = lane >> 4;

  long pair  = (long)blockIdx.x * 8 + wave;
  bool valid = pair < NPAIRS;
  long row0  = (valid ? pair : (NPAIRS - 1)) * 32;
  const float* hrow0 = h   + (row0 + n) * 64;
  const float* hrow1 = h   + (row0 + 16 + n) * 64;
  const float* grow0 = agg + (row0 + n) * 64;
  const float* grow1 = agg + (row0 + 16 + n) * 64;

#pragma unroll
  for (int i = 0; i < 16; ++i) {
    int idx = t + i * 256;
    int c = idx & 63, k = idx >> 6;
    sWA[c * WSTRIDE + k] = gw[idx];
    sWB[c * WSTRIDE + k] = gw[4096 + idx];
  }

  v8f   acc[2][4];
  float bcol[4];
#pragma unroll
  for (int ct = 0; ct < 4; ++ct) {
    bcol[ct] = gb[ct * 16 + n];
#pragma unroll
    for (int i = 0; i < 8; ++i) { acc[0][ct][i] = 0.0f; acc[1][ct][i] = 0.0f; }
  }
  __syncthreads();

#pragma unroll
  for (int kk = 0; kk < 16; ++kk) {
    const int k0 = kk * 4 + kh * 2;
    v2f hv0 = *(const v2f*)(hrow0 + k0);
    v2f hv1 = *(const v2f*)(hrow1 + k0);
    v2f av0 = *(const v2f*)(grow0 + k0);
    v2f av1 = *(const v2f*)(grow1 + k0);
    v2f dv0, dv1;
    dv0.x = av0.x - hv0.x;  dv0.y = av0.y - hv0.y;
    dv1.x = av1.x - hv1.x;  dv1.y = av1.y - hv1.y;
#pragma unroll
    for (int ct = 0; ct < 4; ++ct) {
      v2f bA = *(const v2f*)(&sWA[(ct * 16 + n) * WSTRIDE + k0]);
      v2f bB = *(const v2f*)(&sWB[(ct * 16 + n) * WSTRIDE + k0]);
      acc[0][ct] = wmma_f32(hv0, bA, acc[0][ct]);
      acc[0][ct] = wmma_f32(dv0, bB, acc[0][ct]);
      acc[1][ct] = wmma_f32(hv1, bA, acc[1][ct]);
      acc[1][ct] = wmma_f32(dv1, bB, acc[1][ct]);
    }
  }

  if (valid) {
#pragma unroll
    for (int mt = 0; mt < 2; ++mt)
#pragma unroll
      for (int ct = 0; ct < 4; ++ct)
#pragma unroll
        for (int i = 0; i < 8; ++i) {
          long  r = row0 + mt * 16 + i + 8 * kh;
          float v = acc[mt][ct][i] + bcol[ct];
          out[r * 64 + ct * 16 + n] = v > 0.0f ? v : 0.0f;
        }
  }
}

// ---------------------------------------------------------------------------
// Final FC: out[N,40] = h[N,64] @ W[64,40] + b. 3 column tiles (48); columns
// 40..47 zero-filled in LDS so WMMA runs unguarded; stores masked to col<40.
// ---------------------------------------------------------------------------
__global__ __launch_bounds__(256) void gemm_fc(const float* __restrict__ h,
                                               const float* __restrict__ W,
                                               const float* __restrict__ bias,
                                               float* __restrict__ out) {
  __shared__ float sWt[48 * WSTRIDE];
  const int t    = threadIdx.x;
  const int wave = t >> 5;
  const int lane = t & 31;
  const int n    = lane & 15;
  const int kh   = lane >> 4;

  long pair  = (long)blockIdx.x * 8 + wave;
  bool valid = pair < NPAIRS;
  long row0  = (valid ? pair : (NPAIRS - 1)) * 32;
  const float* arow0 = h + (row0 + n) * 64;
  const float* arow1 = h + (row0 + 16 + n) * 64;

  for (int i = t; i < 48 * WSTRIDE; i += 256) sWt[i] = 0.0f;
  __syncthreads();
  for (int i = t; i < 64 * NCLS; i += 256) {
    int k = i / NCLS, c = i % NCLS;          // W row-major [64][40]
    sWt[c * WSTRIDE + k] = W[i];
  }

  v8f acc[2][3];
#pragma unroll
  for (int ct = 0; ct < 3; ++ct)
#pragma unroll
    for (int i = 0; i < 8; ++i) { acc[0][ct][i] = 0.0f; acc[1][ct][i] = 0.0f; }
  __syncthreads();

#pragma unroll
  for (int kk = 0; kk < 16; ++kk) {
    const int k0 = kk * 4 + kh * 2;
    v2f a0 = *(const v2f*)(arow0 + k0);
    v2f a1 = *(const v2f*)(arow1 + k0);
#pragma unroll
    for (int ct = 0; ct < 3; ++ct) {
      v2f b = *(const v2f*)(&sWt[(ct * 16 + n) * WSTRIDE + k0]);
      acc[0][ct] = wmma_f32(a0, b, acc[0][ct]);
      acc[1][ct] = wmma_f32(a1, b, acc[1][ct]);
    }
  }

  if (valid) {
#pragma unroll
    for (int ct = 0; ct < 3; ++ct) {
      const int col = ct * 16 + n;
      if (col < NCLS) {
        float bv = bias[col];
#pragma unroll
        for (int mt = 0; mt < 2; ++mt)
#pragma unroll
          for (int i = 0; i < 8; ++i) {
            long r = row0 + mt * 16 + i + 8 * kh;
            out[r * 40 + col] = acc[mt][ct][i] + bv;
          }
      }
    }
  }
}

// ---------------------------------------------------------------------------
extern "C" void kernel_launch(void* const* d_in, const int* in_sizes, int n_in,
                              void* d_out, int out_size, void* d_ws, size_t ws_size,
                              hipStream_t stream) {
  (void)in_sizes; (void)n_in; (void)out_size; (void)ws_size;

  const float* x   = (const float*)d_in[0];
  const int*   src = (const int*)d_in[1];
  const int*   dst = (const int*)d_in[2];
  const float* fw0 = (const float*)d_in[3];
  const float* fb0 = (const float*)d_in[4];
  const float* aw0 = (const float*)d_in[5];
  const float* gw0 = (const float*)d_in[6];
  const float* gb0 = (const float*)d_in[7];
  const float* fw1 = (const float*)d_in[8];
  const float* fb1 = (const float*)d_in[9];
  const float* aw1 = (const float*)d_in[10];
  const float* gw1 = (const float*)d_in[11];
  const float* gb1 = (const float*)d_in[12];
  const float* fcw = (const float*)d_in[13];
  const float* fcb = (const float*)d_in[14];

  const long N64 = (long)NNODES * 64;
  float* hA    = (float*)d_ws;       // [N,64]
  float* hB    = hA + N64;           // [N,64]
  float* agg   = hB + N64;           // [N,64]
  float* sdot  = agg + N64;          // [N]
  float* ddot  = sdot + NNODES;      // [N]
  float* emax  = ddot + NNODES;      // [N]
  float* denom = emax + NNODES;      // [N]
  float* ee    = denom + NNODES;     // [E]

  dim3 blk(256);
  dim3 gGemm((NPAIRS + 7) / 8);                 // 391 blocks, 8 wave-pairs each
  dim3 gNode((NNODES + 255) / 256);
  dim3 gEdge((NEDGES + 255) / 256);
  dim3 gAgg((NEDGES + 7) / 8);

  // ---- layer 0 ----
  gemm_feat<<<gGemm, blk, 0, stream>>>(x, fw0, fb0, hA, INDIM);
  attn_dots<<<gNode, blk, 0, stream>>>(hA, aw0, sdot, ddot, emax, denom, agg);
  edge_logits<<<gEdge, blk, 0, stream>>>(src, dst, sdot, ddot, ee, emax);
  edge_exp<<<gEdge, blk, 0, stream>>>(dst, emax, ee, denom);
  edge_agg<<<gAgg, blk, 0, stream>>>(src, dst, ee, denom, hA, agg);
  gemm_update<<<gGemm, blk, 0, stream>>>(hA, agg, gw0, gb0, hB);

  // ---- layer 1 ----
  gemm_feat<<<gGemm, blk, 0, stream>>>(hB, fw1, fb1, hA, HDIM);
  attn_dots<<<gNode, blk, 0, stream>>>(hA, aw1, sdot, ddot, emax, denom, agg);
  edge_logits<<<gEdge, blk, 0, stream>>>(src, dst, sdot, ddot, ee, emax);
  edge_exp<<<gEdge, blk, 0, stream>>>(dst, emax, ee, denom);
  edge_agg<<<gAgg, blk, 0, stream>>>(src, dst, ee, denom, hA, agg);
  gemm_update<<<gGemm, blk, 0, stream>>>(hA, agg, gw1, gb1, hB);

  // ---- classifier ----
  gemm_fc<<<gGemm, blk, 0, stream>>>(hB, fcw, fcb, (float*)d_out);
}